// LogWignerCrystalSlater_52647709114652
// MI455X (gfx1250) — compile-verified
//
#include <hip/hip_runtime.h>
#include <hip/hip_bf16.h>

typedef __attribute__((ext_vector_type(2))) float v2f;
typedef __attribute__((ext_vector_type(8))) float v8f;

#define NMAT     64
#define SLDA     65          // stride-65 rows: conflict-free column access
#define NB       16
#define NTHREADS 256
#define NWAVES   8

#define LBOX     10.0f
#define PI_F     3.14159265358979323846f

__global__ __launch_bounds__(NTHREADS)
void slater_logdet_kernel(const float* __restrict__ rs,
                          const float* __restrict__ log_alpha_p,
                          float* __restrict__ out)
{
    __shared__ float A[NMAT * SLDA];       // 16640 B
    __shared__ float pos[NMAT * 3];        // positions of this spin's particles
    __shared__ float s_piv;
    __shared__ int   s_pidx;
    __shared__ float s_logdet[2];

    const int tid  = threadIdx.x;
    const int lane = tid & 31;
    const int wave = tid >> 5;
    const int b    = blockIdx.x;

    // alpha = clip(exp(log_alpha), 0.5/L^2, 200/L^2)
    float alpha = __expf(log_alpha_p[0]);
    alpha = fminf(fmaxf(alpha, 0.5f / (LBOX * LBOX)), 200.0f / (LBOX * LBOX));

    const float a_lat     = LBOX / 4.0f;       // n=4 since 4^3=64
    const float pi_over_L = PI_F / LBOX;
    const float L_over_pi = LBOX / PI_F;

    for (int spin = 0; spin < 2; ++spin) {
        // ---- load this spin's 64 particle positions into LDS ----
        const float* rbase = rs + ((size_t)b * 128 + (size_t)spin * 64) * 3;
        for (int t = tid; t < NMAT * 3; t += NTHREADS) pos[t] = rbase[t];
        __syncthreads();

        // ---- build Phi[i][j] = exp(-alpha * |L/pi * sin(pi*dr/L)|^2) ----
        const float shift = spin ? (0.5f * a_lat) : 0.0f;
        for (int e = tid; e < NMAT * NMAT; e += NTHREADS) {
            const int i = e >> 6, j = e & 63;
            const int jx = j >> 4, jy = (j >> 2) & 3, jz = j & 3;
            const float cx = (float)jx * a_lat + shift;
            const float cy = (float)jy * a_lat + shift;
            const float cz = (float)jz * a_lat + shift;
            const float sx = L_over_pi * __sinf(pi_over_L * (pos[i * 3 + 0] - cx));
            const float sy = L_over_pi * __sinf(pi_over_L * (pos[i * 3 + 1] - cy));
            const float sz = L_over_pi * __sinf(pi_over_L * (pos[i * 3 + 2] - cz));
            const float r2 = sx * sx + sy * sy + sz * sz;
            A[i * SLDA + j] = __expf(-alpha * r2);
        }
        __syncthreads();

        float logsum = 0.0f;   // meaningful only on tid 0 (wave 0, lane 0)

        // ================= blocked LU with partial pivoting =================
        for (int s = 0; s < 4; ++s) {
            const int k0 = s * NB;

            // ---- panel factorization: cols [k0, k0+16), rows [k, 64) ----
            for (int kk = 0; kk < NB; ++kk) {
                const int k = k0 + kk;

                // pivot argmax over |A[k..63][k]| by wave 0 (shfl reduction)
                if (wave == 0) {
                    const int i0 = lane, i1 = lane + 32;
                    float v0 = (i0 >= k) ? fabsf(A[i0 * SLDA + k]) : -1.0f;
                    float v1 = (i1 >= k) ? fabsf(A[i1 * SLDA + k]) : -1.0f;
                    int   idx = i0; float v = v0;
                    if (v1 > v) { v = v1; idx = i1; }
                    #pragma unroll
                    for (int m = 16; m >= 1; m >>= 1) {
                        const float ov = __shfl_xor(v, m, 32);
                        const int   oi = __shfl_xor(idx, m, 32);
                        if (ov > v) { v = ov; idx = oi; }
                    }
                    if (lane == 0) {
                        const float pv = A[idx * SLDA + k];
                        s_pidx = idx;
                        s_piv  = pv;
                        logsum += __logf(fabsf(pv));   // log|det| = sum log|pivot|
                    }
                }
                __syncthreads();
                const int   pidx = s_pidx;
                const float piv  = s_piv;

                // full row swap k <-> pidx (no-op safe when equal)
                if (tid < NMAT) {
                    const float t0 = A[k * SLDA + tid];
                    A[k * SLDA + tid]    = A[pidx * SLDA + tid];
                    A[pidx * SLDA + tid] = t0;
                }
                __syncthreads();

                // scale multipliers below diagonal
                const float invp = 1.0f / piv;
                if (tid < NMAT && tid > k) A[tid * SLDA + k] *= invp;
                __syncthreads();

                // rank-1 update restricted to panel cols (k, k0+16)
                const int i = k + 1 + (tid >> 2);
                if (i < NMAT) {
                    const float l = A[i * SLDA + k];
                    for (int j = k + 1 + (tid & 3); j < k0 + NB; j += 4)
                        A[i * SLDA + j] -= l * A[k * SLDA + j];
                }
                __syncthreads();
            }

            if (k0 + NB >= NMAT) break;          // last panel: no trailing update
            const int t0c = k0 + NB;
            const int T   = NMAT - t0c;          // 48, 32, 16

            // ---- triangular solve U12 = L11^{-1} * A12, one column/thread ----
            if (tid < T) {
                const int j = t0c + tid;
                float u[NB];
                #pragma unroll
                for (int r = 0; r < NB; ++r) u[r] = A[(k0 + r) * SLDA + j];
                #pragma unroll
                for (int r = 0; r < NB; ++r) {
                    const float ur = u[r];
                    #pragma unroll
                    for (int i2 = r + 1; i2 < NB; ++i2)
                        u[i2] -= A[(k0 + i2) * SLDA + (k0 + r)] * ur;   // broadcast read
                }
                #pragma unroll
                for (int r = 0; r < NB; ++r) A[(k0 + r) * SLDA + j] = u[r];
            }
            __syncthreads();

            // ---- Schur complement: A22 -= L21 * U12 via v_wmma_f32_16x16x4 ----
            const int nbt    = T >> 4;           // 3, 2, 1
            const int ntiles = nbt * nbt;        // 9, 4, 1
            const int half   = lane >> 4;        // lane half (0/1)
            const int l15    = lane & 15;
            for (int t = wave; t < ntiles; t += NWAVES) {
                const int bi = t / nbt, bj = t - bi * nbt;
                const int rb = t0c + bi * 16, cb = t0c + bj * 16;

                // C tile: VGPR v -> row v + 8*half, col = lane&15 (ISA 7.12.2)
                v8f c;
                #pragma unroll
                for (int v = 0; v < 8; ++v)
                    c[v] = A[(rb + v + 8 * half) * SLDA + (cb + l15)];

                #pragma unroll
                for (int kc = 0; kc < NB; kc += 4) {
                    v2f af, bf;
                    #pragma unroll
                    for (int v = 0; v < 2; ++v) {
                        // A 16x4 frag: M = lane&15, K = v + 2*half; negate -> C - L*U
                        af[v] = -A[(rb + l15) * SLDA + (k0 + kc + v + 2 * half)];
                        // B 4x16 frag: K = v + 2*half, N = lane&15
                        bf[v] =  A[(k0 + kc + v + 2 * half) * SLDA + (cb + l15)];
                    }
                    c = __builtin_amdgcn_wmma_f32_16x16x4_f32(
                            false, af, false, bf, (short)0, c, false, false);
                }

                #pragma unroll
                for (int v = 0; v < 8; ++v)
                    A[(rb + v + 8 * half) * SLDA + (cb + l15)] = c[v];
            }
            __syncthreads();
        }

        if (tid == 0) s_logdet[spin] = logsum;
        __syncthreads();
    }

    if (tid == 0) out[b] = s_logdet[0] + s_logdet[1];
}

extern "C" void kernel_launch(void* const* d_in, const int* in_sizes, int n_in,
                              void* d_out, int out_size, void* d_ws, size_t ws_size,
                              hipStream_t stream) {
    (void)in_sizes; (void)n_in; (void)d_ws; (void)ws_size;
    const float* rs        = (const float*)d_in[0];   // (8192, 128, 3) f32
    const float* log_alpha = (const float*)d_in[1];   // scalar f32
    float* out             = (float*)d_out;           // (8192,) f32
    const int batch = out_size;                       // 8192
    slater_logdet_kernel<<<batch, NTHREADS, 0, stream>>>(rs, log_alpha, out);
}